// PPOModelLoss_20401094656264
// MI455X (gfx1250) — compile-verified
//
#include <hip/hip_runtime.h>

// ---------------------------------------------------------------------------
// PPO loss for B=1024, T=4096 (f32). GAE scan done as tiled triangular
// matmul on V_WMMA_F32_16X16X4_F32 (16 batch rows per wave, K=16 via 4
// chained K=4 WMMAs). T split into 32 segments of 128 for parallelism;
// per-segment carries fixed up analytically (constant ratio c = 0.95).
// ---------------------------------------------------------------------------

typedef __attribute__((ext_vector_type(2))) float v2f;
typedef __attribute__((ext_vector_type(8))) float v8f;

#define B_DIM   1024
#define T_DIM   4096
#define SEGLEN  128
#define NSEG    (T_DIM / SEGLEN)     // 32
#define TILES_PER_SEG (SEGLEN / 16)  // 8
#define CLAM    0.95f                // GAMMA * LAM
#define LOGC    (-0.05129329438755058f) // ln(0.95)
#define NTOT_D  ((double)B_DIM * (double)T_DIM)

__device__ __forceinline__ float cpow_small(int e) {
    float r = 1.0f;
    for (int i = 0; i < e; ++i) r *= CLAM;
    return r;
}

// ---------------------------------------------------------------------------
// K1: segment-local GAE suffix scan via WMMA.
// Each wave: 16 rows x one 128-long segment (8 tiles of 16).
// D = W * Dmat : W[i][j] = c^(j-i) (j>=i), Dmat[k][n] = delta(row n, base+k).
// ---------------------------------------------------------------------------
__global__ __launch_bounds__(256) void k1_scan(const float* __restrict__ ov,
                                               const float* __restrict__ rw,
                                               float* __restrict__ adv,
                                               float* __restrict__ Lseg) {
    const int lane = threadIdx.x & 31;
    const int unit = blockIdx.x * 8 + (threadIdx.x >> 5); // 0..2047
    const int rg   = unit & 63;   // row group (16 rows each)
    const int seg  = unit >> 6;   // 0..31
    const int n    = lane & 15;
    const int half = lane >> 4;
    const int row  = rg * 16 + n;
    const size_t rowbase = (size_t)row * T_DIM;

    // Loop-invariant A operands (triangular power matrix, striped per ISA
    // 16x4 f32 A layout: VGPR0={K,K+2}, VGPR1={K+1,K+3} across half-waves).
    v2f a[4];
    #pragma unroll
    for (int j = 0; j < 4; ++j) {
        const int k0 = 4 * j + 2 * half;
        a[j].x = (k0     >= n) ? cpow_small(k0     - n) : 0.0f;
        a[j].y = (k0 + 1 >= n) ? cpow_small(k0 + 1 - n) : 0.0f;
    }
    float cp[8]; // c^(16 - i) for this lane's output time indices i = v + 8*half
    #pragma unroll
    for (int v = 0; v < 8; ++v) cp[v] = cpow_small(16 - (v + 8 * half));

    float carry = 0.0f; // segment-local: adv at first time of previously done tile
    for (int m = TILES_PER_SEG - 1; m >= 0; --m) {
        const int base = seg * SEGLEN + m * 16;

        // Build delta tile as B operand (B 4x16 layout mirrors A striping).
        v2f b[4];
        #pragma unroll
        for (int j = 0; j < 4; ++j) {
            const int t0 = base + 4 * j + 2 * half;
            const int t2 = t0 + 2;
            const float ov0 = ov[rowbase + t0];
            const float ov1 = ov[rowbase + t0 + 1];
            float ov2 = ov[rowbase + (t2 < T_DIM ? t2 : T_DIM - 1)];
            ov2 = (t2 < T_DIM) ? ov2 : 0.0f;       // nextvalue=0 at t=T-1
            const float rw0 = rw[rowbase + t0];
            const float rw1 = rw[rowbase + t0 + 1];
            b[j].x = rw0 + ov1 - ov0;              // GAMMA == 1
            b[j].y = rw1 + ov2 - ov1;
        }

        v8f acc = {0.f, 0.f, 0.f, 0.f, 0.f, 0.f, 0.f, 0.f};
        #pragma unroll
        for (int j = 0; j < 4; ++j)
            acc = __builtin_amdgcn_wmma_f32_16x16x4_f32(
                false, a[j], false, b[j], (short)0, acc, false, false);

        // adv_i = s_i + c^(16-i) * carry
        #pragma unroll
        for (int v = 0; v < 8; ++v) acc[v] = fmaf(cp[v], carry, acc[v]);
        // new carry = adv_0 (held by low half lanes); broadcast to high half
        carry = __shfl(acc[0], n, 32);

        // D layout: lane n holds times base+0..7 (half 0) / base+8..15 (half 1)
        float* dst = adv + rowbase + base + 8 * half;
        *(float4*)(dst)     = make_float4(acc[0], acc[1], acc[2], acc[3]);
        *(float4*)(dst + 4) = make_float4(acc[4], acc[5], acc[6], acc[7]);
    }
    if (half == 0) Lseg[(size_t)row * NSEG + seg] = carry; // local adv at seg start
}

// ---------------------------------------------------------------------------
// K2: per-row scan over 32 segment aggregates. Cseg[row][s] = G_{s+1}
// (true advantage at the first element of segment s+1).
// ---------------------------------------------------------------------------
__global__ void k2_carry(const float* __restrict__ Lseg, float* __restrict__ Cseg) {
    const int row = blockIdx.x * blockDim.x + threadIdx.x;
    if (row >= B_DIM) return;
    const float cseg = expf((float)SEGLEN * LOGC); // c^128
    float g = 0.0f;
    for (int s = NSEG - 1; s >= 0; --s) {
        Cseg[(size_t)row * NSEG + s] = g;
        g = fmaf(cseg, g, Lseg[(size_t)row * NSEG + s]);
    }
}

__device__ __forceinline__ float fixup_adv(const float* __restrict__ advw,
                                           const float* __restrict__ Cseg,
                                           size_t i) {
    const int t   = (int)(i & (T_DIM - 1));
    const size_t row = i >> 12;
    const int seg = t >> 7;
    const float e = (float)((seg + 1) * SEGLEN - t); // in [1,128]
    return fmaf(expf(e * LOGC), Cseg[row * NSEG + seg], advw[i]);
}

__device__ __forceinline__ void block_reduce_add2(float x, float y,
                                                  double* d0, double* d1) {
    __shared__ float r0[256], r1[256];
    const int tid = threadIdx.x;
    r0[tid] = x; r1[tid] = y;
    __syncthreads();
    for (int off = 128; off > 0; off >>= 1) {
        if (tid < off) { r0[tid] += r0[tid + off]; r1[tid] += r1[tid + off]; }
        __syncthreads();
    }
    if (tid == 0) {
        atomicAdd(d0, (double)r0[0]);
        atomicAdd(d1, (double)r1[0]);
    }
}

// ---------------------------------------------------------------------------
// K3: whitening stats: S1 = sum(adv), S2 = sum(adv^2) over all B*T.
// ---------------------------------------------------------------------------
__global__ __launch_bounds__(256) void k3_stats(const float* __restrict__ advw,
                                                const float* __restrict__ Cseg,
                                                double* __restrict__ accd) {
    const size_t total  = (size_t)B_DIM * T_DIM;
    const size_t stride = (size_t)gridDim.x * blockDim.x;
    float s1 = 0.0f, s2 = 0.0f;
    for (size_t i = (size_t)blockIdx.x * blockDim.x + threadIdx.x; i < total; i += stride) {
        const float a = fixup_adv(advw, Cseg, i);
        s1 += a;
        s2 = fmaf(a, a, s2);
    }
    block_reduce_add2(s1, s2, &accd[0], &accd[1]);
}

// ---------------------------------------------------------------------------
// K4: clipped PG + VF loss numerator and mask count.
// ---------------------------------------------------------------------------
__global__ __launch_bounds__(256) void k4_loss(const float* __restrict__ lp,
                                               const float* __restrict__ val,
                                               const float* __restrict__ olp,
                                               const float* __restrict__ ov,
                                               const int*   __restrict__ mask,
                                               const float* __restrict__ advw,
                                               const float* __restrict__ Cseg,
                                               double* __restrict__ accd) {
    const double s1 = accd[0], s2 = accd[1];
    const float mean = (float)(s1 * (1.0 / NTOT_D));
    const float var  = (float)((s2 - s1 * s1 * (1.0 / NTOT_D)) * (1.0 / (NTOT_D - 1.0)));
    const float rstd = rsqrtf(var + 1e-8f);

    const size_t total  = (size_t)B_DIM * T_DIM;
    const size_t stride = (size_t)gridDim.x * blockDim.x;
    float lsum = 0.0f, nsum = 0.0f;
    for (size_t i = (size_t)blockIdx.x * blockDim.x + threadIdx.x; i < total; i += stride) {
        const float araw = fixup_adv(advw, Cseg, i);
        const float ovv  = ov[i];
        const float ret  = araw + ovv;              // returns (pre-whitening adv)
        const float aw   = (araw - mean) * rstd;    // whitened advantage

        const float vv = val[i];
        const float vc = fminf(fmaxf(vv, ovv - 0.2f), ovv + 0.2f);
        const float d1 = vv - ret, d2 = vc - ret;
        const float vf = fmaxf(d1 * d1, d2 * d2);

        const float mf = (float)mask[i];
        const float ratio = expf((lp[i] - olp[i]) * mf);
        const float rc = fminf(fmaxf(ratio, 0.8f), 1.2f);
        const float pg = fmaxf(-aw * ratio, -aw * rc);

        lsum += (pg + 0.5f * vf) * mf;
        nsum += mf;
    }
    block_reduce_add2(lsum, nsum, &accd[2], &accd[3]);
}

__global__ void k5_final(const double* __restrict__ accd, float* __restrict__ out) {
    out[0] = (float)(accd[2] / accd[3]);
}

// ---------------------------------------------------------------------------
// Workspace layout: [0,64)   : 4 doubles (S1, S2, loss_num, n_mask)
//                   [1K,513K): Lseg  (B*NSEG floats)
//                   [513K..) : Cseg  (B*NSEG floats)
//                   [1M+1K..): advw  (B*T floats, 16 MB)
// ---------------------------------------------------------------------------
extern "C" void kernel_launch(void* const* d_in, const int* in_sizes, int n_in,
                              void* d_out, int out_size, void* d_ws, size_t ws_size,
                              hipStream_t stream) {
    (void)in_sizes; (void)n_in; (void)out_size; (void)ws_size;
    const float* logprobs = (const float*)d_in[0];
    const float* values   = (const float*)d_in[1];
    const float* old_logp = (const float*)d_in[2];
    const float* old_vals = (const float*)d_in[3];
    const float* rewards  = (const float*)d_in[4];
    const int*   mask     = (const int*)d_in[5];

    char* ws = (char*)d_ws;
    double* accd = (double*)ws;
    float* Lseg  = (float*)(ws + 1024);
    float* Cseg  = (float*)(ws + 1024 + 512 * 1024);
    float* advw  = (float*)(ws + 1024 + 1024 * 1024);

    hipMemsetAsync(accd, 0, 4 * sizeof(double), stream);
    k1_scan<<<256, 256, 0, stream>>>(old_vals, rewards, advw, Lseg);  // 2048 waves
    k2_carry<<<4, 256, 0, stream>>>(Lseg, Cseg);
    k3_stats<<<1024, 256, 0, stream>>>(advw, Cseg, accd);
    k4_loss<<<1024, 256, 0, stream>>>(logprobs, values, old_logp, old_vals, mask,
                                      advw, Cseg, accd);
    k5_final<<<1, 1, 0, stream>>>(accd, (float*)d_out);
}